// VQCNN_Parameterized_12756052869709
// MI455X (gfx1250) — compile-verified
//
#include <hip/hip_runtime.h>
#include <math.h>

// ---------------------------------------------------------------------------
// VQC (8 qubits, 10 CRX gates) + tiny MLP head.
// Kernel 1: one wave32 per batch element. 256 complex amplitudes live in
//           registers: lane = index bits 7..3 (qubits 0-4), 8 complex values
//           per lane indexed by bits 2..0 (qubits 5-7). Cross-lane gates use
//           shfl_xor lane permutes; in-lane gates are register updates.
//           Control bits are folded into per-lane (ce,se) coefficients so
//           EXEC stays full (no divergent branches). Hardware v_sin/v_cos
//           (args <= pi, well inside accurate range). Writes z=(<Z3>,<Z7>).
// Kernel 2: MLP head via V_WMMA_F32_16X16X4_F32 (z(16x2 pad 4) @ w1(2x10 pad
//           4x16)), tanh, then K=10 dot with w2 via 16-lane shfl reduction.
// ---------------------------------------------------------------------------

typedef float v2f __attribute__((ext_vector_type(2)));
typedef float v8f __attribute__((ext_vector_type(8)));

// CRX pair update with effective coeffs: a' = ce*a - i*se*b ; b' = ce*b - i*se*a
#define PAIR_UPD(CE, SE, A, B_)                                         \
  { float ar = re[A], ai = im[A], br = re[B_], bi = im[B_];             \
    re[A]  = fmaf((CE), ar,  (SE) * bi);                                \
    im[A]  = fmaf((CE), ai, -(SE) * br);                                \
    re[B_] = fmaf((CE), br,  (SE) * ai);                                \
    im[B_] = fmaf((CE), bi, -(SE) * ar); }

// Cross-lane CRX gate, control = lane bit (shared by both shuffle partners).
// ce=1,se=0 for uncontrolled lanes -> unconditional update, EXEC stays full.
#define XGATE_LANECTRL(G, CSHIFT, LMASK)                                \
  { float ct = __shfl(tc, (G), 32), st = __shfl(ts, (G), 32);           \
    const bool ctl = (lane >> (CSHIFT)) & 1;                            \
    const float ce = ctl ? ct : 1.0f;                                   \
    const float se = ctl ? st : 0.0f;                                   \
    _Pragma("unroll") for (int r = 0; r < 8; ++r) {                     \
      float pre = __shfl_xor(re[r], (LMASK), 32);                       \
      float pim = __shfl_xor(im[r], (LMASK), 32);                       \
      float nr = fmaf(ce, re[r],  se * pim);                            \
      float ni = fmaf(ce, im[r], -se * pre);                            \
      re[r] = nr; im[r] = ni;                                           \
    }                                                                   \
  }

__global__ __launch_bounds__(256)
void vqc_state_kernel(const float* __restrict__ x,
                      const float* __restrict__ theta,
                      float* __restrict__ zbuf, int B)
{
  const int lane = threadIdx.x & 31;
  const int b    = blockIdx.x * (blockDim.x >> 5) + (threadIdx.x >> 5);
  if (b >= B) return;                       // wave-uniform exit

  // ---- load the 8 angles (one lane each) and broadcast -------------------
  float xv = 0.f;
  if (lane < 8) xv = x[b * 8 + lane];
  float ch[8], sh[8];
  #pragma unroll
  for (int q = 0; q < 8; ++q) {
    float h = 0.5f * __shfl(xv, q, 32);
    ch[q] = __cosf(h);                      // |h| <= pi: v_cos_f32 accurate
    sh[q] = __sinf(h);
  }

  // ---- per-gate cos/sin(theta/2), one lane each, broadcast at use --------
  float tc = 1.f, ts = 0.f;
  if (lane < 10) {
    float th = 0.5f * theta[lane];
    tc = __cosf(th);
    ts = __sinf(th);
  }

  // ---- encode: amplitude(i) = prod_q f_q(bit_q), bit_q = (i >> (7-q)) & 1
  float re[8], im[8];
  {
    float lp = ((lane >> 4) & 1 ? sh[0] : ch[0])
             * ((lane >> 3) & 1 ? sh[1] : ch[1])
             * ((lane >> 2) & 1 ? sh[2] : ch[2])
             * ((lane >> 1) & 1 ? sh[3] : ch[3])
             * ((lane      ) & 1 ? sh[4] : ch[4]);
    #pragma unroll
    for (int r = 0; r < 8; ++r) {
      float rp = ((r >> 2) & 1 ? sh[5] : ch[5])
               * ((r >> 1) & 1 ? sh[6] : ch[6])
               * ((r      ) & 1 ? sh[7] : ch[7]);
      re[r] = lp * rp;
      im[r] = 0.f;
    }
  }

  // ---- gates: WIRES = (0,1)(1,2)(2,3)(3,4)(4,5)(5,6)(6,7)(7,0)(0,1)(1,2) -
  XGATE_LANECTRL(0, 4, 8)                       // (c=0,t=1)
  XGATE_LANECTRL(1, 3, 4)                       // (1,2)
  XGATE_LANECTRL(2, 2, 2)                       // (2,3)
  XGATE_LANECTRL(3, 1, 1)                       // (3,4)
  { // (4,5): ctrl = lane bit 0, target = register bit 2
    float ct = __shfl(tc, 4, 32), st = __shfl(ts, 4, 32);
    const float ce = (lane & 1) ? ct : 1.0f;
    const float se = (lane & 1) ? st : 0.0f;
    PAIR_UPD(ce, se, 0, 4) PAIR_UPD(ce, se, 1, 5)
    PAIR_UPD(ce, se, 2, 6) PAIR_UPD(ce, se, 3, 7)
  }
  { // (5,6): ctrl = r bit 2, target = r bit 1 -> pairs (4,6),(5,7)
    float ct = __shfl(tc, 5, 32), st = __shfl(ts, 5, 32);
    PAIR_UPD(ct, st, 4, 6) PAIR_UPD(ct, st, 5, 7)
  }
  { // (6,7): ctrl = r bit 1, target = r bit 0 -> pairs (2,3),(6,7)
    float ct = __shfl(tc, 6, 32), st = __shfl(ts, 6, 32);
    PAIR_UPD(ct, st, 2, 3) PAIR_UPD(ct, st, 6, 7)
  }
  { // (7,0): ctrl = r bit 0 (odd r only), target = lane bit 4
    float ct = __shfl(tc, 7, 32), st = __shfl(ts, 7, 32);
    #pragma unroll
    for (int r = 1; r < 8; r += 2) {
      float pre = __shfl_xor(re[r], 16, 32);
      float pim = __shfl_xor(im[r], 16, 32);
      float nr = fmaf(ct, re[r],  st * pim);
      float ni = fmaf(ct, im[r], -st * pre);
      re[r] = nr; im[r] = ni;
    }
  }
  XGATE_LANECTRL(8, 4, 8)                       // (0,1) again
  XGATE_LANECTRL(9, 3, 4)                       // (1,2) again

  // ---- <Z3> (lane bit 1) and <Z7> (r bit 0) ------------------------------
  float p[8];
  #pragma unroll
  for (int r = 0; r < 8; ++r) p[r] = fmaf(re[r], re[r], im[r] * im[r]);
  float sall = ((p[0] + p[1]) + (p[2] + p[3])) + ((p[4] + p[5]) + (p[6] + p[7]));
  float s7   = ((p[0] - p[1]) + (p[2] - p[3])) + ((p[4] - p[5]) + (p[6] - p[7]));
  float z3 = ((lane >> 1) & 1) ? -sall : sall;
  float z7 = s7;
  #pragma unroll
  for (int m = 16; m >= 1; m >>= 1) {
    z3 += __shfl_xor(z3, m, 32);
    z7 += __shfl_xor(z7, m, 32);
  }
  if (lane == 0) {
    zbuf[2 * b + 0] = z3;
    zbuf[2 * b + 1] = z7;
  }
}

// ---------------------------------------------------------------------------
// MLP head: h = tanh(z @ w1 + b1); out = sigmoid(h @ w2 + b2)
// One wave per 16 rows; layer 1 via V_WMMA_F32_16X16X4_F32 (K 2->4, N 10->16).
// ---------------------------------------------------------------------------
__global__ __launch_bounds__(256)
void vqc_mlp_wmma_kernel(const float* __restrict__ zbuf,
                         const float* __restrict__ w1,
                         const float* __restrict__ b1,
                         const float* __restrict__ w2,
                         const float* __restrict__ b2,
                         float* __restrict__ out, int B)
{
  const int lane = threadIdx.x & 31;
  const int tile = blockIdx.x * (blockDim.x >> 5) + (threadIdx.x >> 5);
  const int row0 = tile * 16;
  if (row0 >= B) return;                     // wave-uniform; EXEC stays full

  // A (16x4): lanes 0-15 carry K0=z3, K1=z7; lanes 16-31 carry K2=K3=0
  v2f a; a.x = 0.f; a.y = 0.f;
  if (lane < 16 && (row0 + lane) < B) {
    a.x = zbuf[2 * (row0 + lane) + 0];
    a.y = zbuf[2 * (row0 + lane) + 1];
  }
  // B (4x16): lanes 0-15 carry K0=w1[0][n], K1=w1[1][n] (n<10), rest zero
  v2f bm; bm.x = 0.f; bm.y = 0.f;
  if (lane < 10) { bm.x = w1[lane]; bm.y = w1[10 + lane]; }

  v8f c = {};
  c = __builtin_amdgcn_wmma_f32_16x16x4_f32(
        /*neg_a=*/false, a, /*neg_b=*/false, bm,
        /*c_mod=*/(short)0, c, /*reuse_a=*/false, /*reuse_b=*/false);

  // D layout: VGPR j, lanes 0-15 -> (M=j, N=lane); lanes 16-31 -> (M=j+8)
  const int n     = lane & 15;
  const float b1n = (n < 10) ? b1[n] : 0.f;
  const float w2n = (n < 10) ? w2[n] : 0.f;
  const float bias2 = b2[0];
  #pragma unroll
  for (int j = 0; j < 8; ++j) {
    float h   = tanhf(c[j] + b1n);
    float acc = h * w2n;
    acc += __shfl_xor(acc, 1, 32);
    acc += __shfl_xor(acc, 2, 32);
    acc += __shfl_xor(acc, 4, 32);
    acc += __shfl_xor(acc, 8, 32);           // stays within 16-lane half
    if (n == 0) {
      int m = row0 + j + ((lane >= 16) ? 8 : 0);
      if (m < B) out[m] = 1.f / (1.f + expf(-(acc + bias2)));
    }
  }
}

extern "C" void kernel_launch(void* const* d_in, const int* in_sizes, int n_in,
                              void* d_out, int out_size, void* d_ws, size_t ws_size,
                              hipStream_t stream)
{
  const float* x     = (const float*)d_in[0];
  const float* theta = (const float*)d_in[1];
  const float* w1    = (const float*)d_in[2];
  const float* b1    = (const float*)d_in[3];
  const float* w2    = (const float*)d_in[4];
  const float* b2    = (const float*)d_in[5];
  float* out = (float*)d_out;
  const int B = in_sizes[0] / 8;

  float* zbuf = (float*)d_ws;                // B*2 floats (512 KB for B=65536)

  // Kernel 1: 8 waves/block, one batch element per wave.
  const int wpb1 = 8;
  const int blocks1 = (B + wpb1 - 1) / wpb1;
  vqc_state_kernel<<<blocks1, 32 * wpb1, 0, stream>>>(x, theta, zbuf, B);

  // Kernel 2: 16 rows per wave, 8 waves/block.
  const int tiles  = (B + 15) / 16;
  const int blocks2 = (tiles + 7) / 8;
  vqc_mlp_wmma_kernel<<<blocks2, 256, 0, stream>>>(zbuf, w1, b1, w2, b2, out, B);
}